// TransducerLoss_13606456394349
// MI455X (gfx1250) — compile-verified
//
#include <hip/hip_runtime.h>
#include <cstdint>
#include <cstddef>

#define AS1 __attribute__((address_space(1)))
#define AS3 __attribute__((address_space(3)))

#ifndef __has_builtin
#define __has_builtin(x) 0
#endif

// ---------------- CDNA5 async global->LDS copy (gfx1250) ----------------
__device__ __forceinline__ void async_g2l_b32(const float* g, float* l) {
#if __has_builtin(__builtin_amdgcn_global_load_async_to_lds_b32)
  AS1 void* gv = (AS1 void*)(void*)g;   // addrspacecast (accepted last round)
  AS3 void* lv = (AS3 void*)(void*)l;
  __builtin_amdgcn_global_load_async_to_lds_b32((AS1 int*)gv, (AS3 int*)lv,
                                                /*imm offset*/0, /*cpol*/0);
#else
  unsigned lds_off = (unsigned)(size_t)(AS3 void*)(void*)l;
  asm volatile("global_load_async_to_lds_b32 %0, %1, off"
               :: "v"(lds_off), "v"(g) : "memory");
#endif
}

__device__ __forceinline__ void wait_async_le4() {
#if __has_builtin(__builtin_amdgcn_s_wait_asynccnt)
  __builtin_amdgcn_s_wait_asynccnt(4);
#else
  asm volatile("s_wait_asynccnt 4" ::: "memory");
#endif
  asm volatile("" ::: "memory");  // compiler fence: keep LDS reads below the wait
}

// ---------------- problem constants (from reference setup) ----------------
constexpr int TT = 256;
constexpr int UU = 128;   // block size = U
constexpr int VV = 256;   // blank = VV-1
constexpr float NEGF = -1e30f;

// alpha[t,u] = c[u] + LSE_{k<=u}(A[k]-c[k]),  A = alpha[t-1]+blank[t-1],
// c = exclusive cumsum over u of emit[t].
__global__ __launch_bounds__(UU) void rnnt_alpha(const float* __restrict__ lp,
                                                 const int*   __restrict__ labels,
                                                 const int*   __restrict__ lengths,
                                                 const int*   __restrict__ lablens,
                                                 float*       __restrict__ per_n) {
  const int n    = blockIdx.x;
  const int u    = threadIdx.x;
  const int lane = u & 31;
  const int w    = u >> 5;           // wave id 0..3

  __shared__ float ebuf[3][UU];      // staged emit log-probs (triple buffered)
  __shared__ float bbuf[3][UU];      // staged blank log-probs
  __shared__ float wsum[4];          // per-wave cumsum totals
  __shared__ float wm[4];            // per-wave LSE (m) totals
  __shared__ float wss[4];           // per-wave LSE (s) totals

  const float* base = lp + (size_t)n * TT * UU * VV;
  const int lab = (u < UU - 1) ? labels[n * (UU - 1) + u] : (VV - 1);
  const int myT = lengths[n];        // uniform per block
  const int myU = lablens[n];

  auto issue = [&](int t, int slot) {
    const float* r = base + ((size_t)t * UU + u) * VV;
    async_g2l_b32(r + lab,      &ebuf[slot][u]);
    async_g2l_b32(r + (VV - 1), &bbuf[slot][u]);
  };

  // 2-deep pipeline prologue
  issue(0, 0);
  issue(1, 1);

  float A = (u == 0) ? 0.0f : NEGF;  // A0 one-hot at u=0

  for (int t = 0; t < myT; ++t) {
    int tn = t + 2;
    if (tn >= TT) tn -= TT;          // harmless wraparound reload, keeps wait imm fixed
    issue(tn, (t + 2) % 3);
    wait_async_le4();                // oldest pair (slot t%3) complete (in-order)

    const int slot = t % 3;
    float e = (u < UU - 1) ? ebuf[slot][u] : 0.0f;
    float b = bbuf[slot][u];

    // ---- exclusive prefix sum c[u] of emit over u ----
    float incl = e;
#pragma unroll
    for (int d = 1; d < 32; d <<= 1) {
      float y = __shfl_up(incl, (unsigned)d, 32);
      if (lane >= d) incl += y;
    }
    if (lane == 31) wsum[w] = incl;
    __syncthreads();
    float wp = 0.0f;
    for (int k = 0; k < w; ++k) wp += wsum[k];
    float up1 = __shfl_up(incl, 1u, 32);
    float c = wp + ((lane == 0) ? 0.0f : up1);

    // ---- inclusive cum-logsumexp scan of x[u] = A[u] - c[u] ----
    // value represented as (m, s): LSE = m + log(s); s >= 1 always.
    float m = A - c;
    float s = 1.0f;
#pragma unroll
    for (int d = 1; d < 32; d <<= 1) {
      float m2 = __shfl_up(m, (unsigned)d, 32);
      float s2 = __shfl_up(s, (unsigned)d, 32);
      if (lane >= d) {
        float M = fmaxf(m, m2);
        s = s * expf(m - M) + s2 * expf(m2 - M);
        m = M;
      }
    }
    if (lane == 31) { wm[w] = m; wss[w] = s; }
    __syncthreads();
    float Mp = NEGF, Sp = 0.0f;
    for (int k = 0; k < w; ++k) {
      float M2 = wm[k], S2 = wss[k];
      float M = fmaxf(Mp, M2);
      Sp = Sp * expf(Mp - M) + S2 * expf(M2 - M);
      Mp = M;
    }
    {
      float M = fmaxf(Mp, m);
      s = Sp * expf(Mp - M) + s * expf(m - M);
      m = M;
    }
    float alpha = c + m + logf(s);

    if (t == myT - 1 && u == myU) per_n[n] = -(alpha + b);  // -loglik
    A = alpha + b;                    // next row's A = alpha[t] + blank[t]
  }
}

// fixed-order reduction -> deterministic across graph replays
__global__ void rnnt_reduce(const float* __restrict__ per_n,
                            float* __restrict__ out, int N) {
  if (threadIdx.x == 0 && blockIdx.x == 0) {
    float acc = 0.0f;
    for (int i = 0; i < N; ++i) acc += per_n[i];
    out[0] = acc / (float)N;
  }
}

extern "C" void kernel_launch(void* const* d_in, const int* in_sizes, int n_in,
                              void* d_out, int out_size, void* d_ws, size_t ws_size,
                              hipStream_t stream) {
  const float* lp      = (const float*)d_in[0];
  const int*   labels  = (const int*)d_in[1];
  const int*   lengths = (const int*)d_in[2];
  const int*   lablens = (const int*)d_in[3];
  const int    N       = in_sizes[2];

  float* per_n = (float*)d_ws;
  rnnt_alpha<<<N, UU, 0, stream>>>(lp, labels, lengths, lablens, per_n);
  rnnt_reduce<<<1, 32, 0, stream>>>(per_n, (float*)d_out, N);
}